// PointerNetwork_18021682774823
// MI455X (gfx1250) — compile-verified
//
#include <hip/hip_runtime.h>
#include <hip/hip_bf16.h>
#include <math.h>

// ---------------------------------------------------------------------------
// Problem constants (match reference: B=4, D=128, E=512, H=512)
// ---------------------------------------------------------------------------
#define PN_B 4
#define PN_D 128
#define PN_E 512
#define PN_H 512

typedef __attribute__((ext_vector_type(2))) float v2f;
typedef __attribute__((ext_vector_type(8))) float v8f;

// ---------------------------------------------------------------------------
// Fast device math (CDNA5 TRANS pipe)
// ---------------------------------------------------------------------------
__device__ __forceinline__ float fexp2(float x)  { return __builtin_amdgcn_exp2f(x); }
__device__ __forceinline__ float flog2(float x)  { return __builtin_amdgcn_logf(x); }
__device__ __forceinline__ float frcp(float x)   { return __builtin_amdgcn_rcpf(x); }

__device__ __forceinline__ float fast_tanh(float x) {
#if __has_builtin(__builtin_amdgcn_tanhf)
    return __builtin_amdgcn_tanhf(x);             // v_tanh_f32
#else
    // tanh(x) = (1 - e^{-2x}) / (1 + e^{-2x}); clamp so exp2 never overflows
    float xc = fminf(fmaxf(x, -15.0f), 15.0f);
    float t  = fexp2(xc * -2.8853900817779268f);  // e^{-2x} = 2^{-2x*log2(e)}
    return (1.0f - t) * frcp(1.0f + t);           // v_exp_f32 + v_rcp_f32
#endif
}

// ---------------------------------------------------------------------------
// Kernel 1: Y[m,o] = sum_h X[m,h] * W[o,h]     (X:(M,H) rm, W:(O,H) rm)
// One wave (32 lanes) owns a 16x16 tile of Y; K-loop of v_wmma_f32_16x16x4_f32.
//
// A-fragment layout (16x4 f32, ISA 7.12.2): lane l<16 -> M=l, VGPR{0,1}=K{0,1};
// lanes 16..31 -> M=l-16, VGPR{0,1}=K{2,3}. Row-major X makes this a float2 load.
// B-fragment (4x16) mirrors it with N across lanes; W row-major in h -> float2.
// C/D: lane l<16 -> rows M=0..7 at col l; lanes 16..31 -> rows 8..15.
// ---------------------------------------------------------------------------
__global__ __launch_bounds__(32)
void pn_gemm_xwt_wmma(const float* __restrict__ X,
                      const float* __restrict__ W,
                      float* __restrict__ Y,
                      int M, int H, int O)
{
    const int lane = threadIdx.x;        // 0..31, EXEC all ones (no divergence)
    const int half = lane >> 4;          // 0 or 1
    const int l    = lane & 15;
    const int m0   = blockIdx.x * 16;
    const int o0   = blockIdx.y * 16;

    const float* xrow = X + (size_t)(m0 + l) * H + 2 * half;
    const float* wrow = W + (size_t)(o0 + l) * H + 2 * half;

    v8f c0 = {};
    v8f c1 = {};
    #pragma unroll 4
    for (int k = 0; k < H; k += 8) {
        v2f a0 = *(const v2f*)(xrow + k);
        v2f b0 = *(const v2f*)(wrow + k);
        v2f a1 = *(const v2f*)(xrow + k + 4);
        v2f b1 = *(const v2f*)(wrow + k + 4);
        c0 = __builtin_amdgcn_wmma_f32_16x16x4_f32(false, a0, false, b0,
                                                   (short)0, c0, false, false);
        c1 = __builtin_amdgcn_wmma_f32_16x16x4_f32(false, a1, false, b1,
                                                   (short)0, c1, false, false);
    }
    v8f c = c0 + c1;

    float* ybase = Y + (size_t)(m0 + 8 * half) * O + (o0 + l);
    #pragma unroll
    for (int r = 0; r < 8; ++r)
        ybase[(size_t)r * O] = c[r];
}

// ---------------------------------------------------------------------------
// Kernel 2: fused additive-attention scores + mask + log_softmax.
// One block per (b,d): 256 threads, each computes E/256 = 2 scores:
//   s[e] = sum_h tanh(dec[h] + enc[b,e,h]) * v[h]
// then block-local max / logsumexp over E in LDS, write log_softmax.
// dec row + v staged in LDS (reused 512x / 512x per block); enc streamed
// from L2 via b128 loads.
// ---------------------------------------------------------------------------
__global__ __launch_bounds__(256)
void pn_attn_logsoftmax(const float* __restrict__ enc_t,   // (B,E,H)
                        const float* __restrict__ dec_t,   // (B,D,H)
                        const unsigned char* __restrict__ mask, // (B,E) bool
                        const float* __restrict__ v,       // (H)
                        float* __restrict__ out)           // (B,D,E)
{
    __shared__ float s_dec[PN_H];
    __shared__ float s_v[PN_H];
    __shared__ float s_scores[PN_E];
    __shared__ float s_red[256];

    const int bd = blockIdx.x;           // 0 .. B*D-1
    const int b  = bd / PN_D;
    const int t  = threadIdx.x;

    // stage dec row and v
    const float* dec = dec_t + (size_t)bd * PN_H;
    for (int i = t; i < PN_H; i += 256) {
        s_dec[i] = dec[i];
        s_v[i]   = v[i];
    }
    __syncthreads();

    const float4* dec4 = (const float4*)s_dec;
    const float4* v4   = (const float4*)s_v;

    // each thread: scores for e = t and e = t + 256
    #pragma unroll
    for (int j = 0; j < PN_E; j += 256) {
        const int e = j + t;
        const float4* enc4 = (const float4*)(enc_t + ((size_t)b * PN_E + e) * PN_H);
        float a0 = 0.f, a1 = 0.f, a2 = 0.f, a3 = 0.f;
        #pragma unroll 4
        for (int h4 = 0; h4 < PN_H / 4; ++h4) {
            float4 ev = enc4[h4];        // global_load_b128 (L2-resident)
            float4 dv = dec4[h4];        // ds_load_b128
            float4 vv = v4[h4];          // ds_load_b128
            a0 += fast_tanh(dv.x + ev.x) * vv.x;
            a1 += fast_tanh(dv.y + ev.y) * vv.y;
            a2 += fast_tanh(dv.z + ev.z) * vv.z;
            a3 += fast_tanh(dv.w + ev.w) * vv.w;
        }
        float s = (a0 + a1) + (a2 + a3);
        if (mask[b * PN_E + e]) s = -INFINITY;
        s_scores[e] = s;
    }
    __syncthreads();

    // block max over E
    float m = fmaxf(s_scores[t], s_scores[t + 256]);
    s_red[t] = m;
    __syncthreads();
    #pragma unroll
    for (int s = 128; s > 0; s >>= 1) {
        if (t < s) s_red[t] = fmaxf(s_red[t], s_red[t + s]);
        __syncthreads();
    }
    const float gmax = s_red[0];
    __syncthreads();

    // block sum of exp(s - gmax)
    const float L2E = 1.4426950408889634f;
    float sum = fexp2((s_scores[t] - gmax) * L2E) +
                fexp2((s_scores[t + 256] - gmax) * L2E);
    s_red[t] = sum;
    __syncthreads();
    #pragma unroll
    for (int s = 128; s > 0; s >>= 1) {
        if (t < s) s_red[t] = s_red[t] + s_red[t + s];
        __syncthreads();
    }
    const float lse = gmax + flog2(s_red[0]) * 0.6931471805599453f;

    float* orow = out + (size_t)bd * PN_E;
    orow[t]       = s_scores[t]       - lse;
    orow[t + 256] = s_scores[t + 256] - lse;
}

// ---------------------------------------------------------------------------
// Launch
// ---------------------------------------------------------------------------
extern "C" void kernel_launch(void* const* d_in, const int* in_sizes, int n_in,
                              void* d_out, int out_size, void* d_ws, size_t ws_size,
                              hipStream_t stream) {
    const float*         x_decoder = (const float*)d_in[0];         // (B,D,H)
    const float*         x_encoder = (const float*)d_in[1];         // (B,E,H)
    const unsigned char* mask      = (const unsigned char*)d_in[2]; // (B,E) bool
    const float*         W1        = (const float*)d_in[3];         // (H,H)
    const float*         W2        = (const float*)d_in[4];         // (H,H)
    const float*         v         = (const float*)d_in[5];         // (H)
    float*               out       = (float*)d_out;                 // (B,D,E)

    // workspace: enc_t (B*E*H floats = 4MB) | dec_t (B*D*H floats = 1MB)
    float* enc_t = (float*)d_ws;
    float* dec_t = enc_t + (size_t)PN_B * PN_E * PN_H;

    // enc_t[b,e,o] = sum_h x_encoder[b,e,h] * W1[o,h]   -> M = B*E = 2048
    {
        dim3 grid((PN_B * PN_E) / 16, PN_H / 16);
        pn_gemm_xwt_wmma<<<grid, 32, 0, stream>>>(x_encoder, W1, enc_t,
                                                  PN_B * PN_E, PN_H, PN_H);
    }
    // dec_t[b,d,o] = sum_h x_decoder[b,d,h] * W2[o,h]   -> M = B*D = 512
    {
        dim3 grid((PN_B * PN_D) / 16, PN_H / 16);
        pn_gemm_xwt_wmma<<<grid, 32, 0, stream>>>(x_decoder, W2, dec_t,
                                                  PN_B * PN_D, PN_H, PN_H);
    }
    // fused tanh-attention + mask + log_softmax
    {
        dim3 grid(PN_B * PN_D);
        pn_attn_logsoftmax<<<grid, 256, 0, stream>>>(enc_t, dec_t, mask, v, out);
    }
}